// InvariantAttention_43147241456122
// MI455X (gfx1250) — compile-verified
//
#include <hip/hip_runtime.h>
#include <hip/hip_bf16.h>
#include <math.h>

#define B_ 4
#define T_ 512
#define H_ 1024
#define D_ 32
#define NH_ 32
#define E_ 256
#define S_ 768
#define TBLK 128
#define EPS_ 1e-5f
#define CUTOFF_ 1e-5f
#define SCALING_ 0.17677669529663687f

typedef __attribute__((ext_vector_type(2))) float v2f;
typedef __attribute__((ext_vector_type(8))) float v8f;

__device__ __forceinline__ v8f wmma_f32(v2f a, v2f b, v8f c) {
  // V_WMMA_F32_16X16X4_F32 : D(16x16,f32) = A(16x4,f32) x B(4x16,f32) + C
  return __builtin_amdgcn_wmma_f32_16x16x4_f32(false, a, false, b, (short)0, c,
                                               false, false);
}

// DPP16 row-rotate move (rotation stays inside each 16-lane row, which is
// exactly the half-wave domain of the WMMA C layout).
template <int CTRL>
__device__ __forceinline__ float dpp_mov(float x) {
  int i = __float_as_int(x);
  return __int_as_float(
      __builtin_amdgcn_update_dpp(i, i, CTRL, 0xF, 0xF, false));
}

// Reduction over the 16 lanes of a row via rotate-combine (all lanes end with
// the full result). ROW_ROR:n = 0x120 + n.
__device__ __forceinline__ float row_max16(float x) {
  x = fmaxf(x, dpp_mov<0x121>(x));
  x = fmaxf(x, dpp_mov<0x122>(x));
  x = fmaxf(x, dpp_mov<0x124>(x));
  x = fmaxf(x, dpp_mov<0x128>(x));
  return x;
}
__device__ __forceinline__ float row_sum16(float x) {
  x += dpp_mov<0x121>(x);
  x += dpp_mov<0x122>(x);
  x += dpp_mov<0x124>(x);
  x += dpp_mov<0x128>(x);
  return x;
}

// ---------------------------------------------------------------------------
// Kernel 1: PBC gather-expansion of K/V  ->  kexp/vexp [B, S, H]
// ---------------------------------------------------------------------------
__global__ __launch_bounds__(256) void expand_kv_kernel(
    const float* __restrict__ k, const float* __restrict__ v,
    const int* __restrict__ outcell,
    float* __restrict__ kexp, float* __restrict__ vexp)
{
  int b = blockIdx.x / S_;
  int s = blockIdx.x % S_;
  int src = (s < T_) ? s : outcell[b * E_ + (s - T_)];
  const float4* ks = (const float4*)(k + ((size_t)b * T_ + src) * H_);
  const float4* vs = (const float4*)(v + ((size_t)b * T_ + src) * H_);
  float4* kd = (float4*)(kexp + ((size_t)b * S_ + s) * H_);
  float4* vd = (float4*)(vexp + ((size_t)b * S_ + s) * H_);
  kd[threadIdx.x] = ks[threadIdx.x];   // 256 threads * float4 == H
  vd[threadIdx.x] = vs[threadIdx.x];
}

// ---------------------------------------------------------------------------
// Kernel 2: fused flash attention.
// Grid: (b, h, t-block of 128). 8 waves; wave w owns query rows [w*16, w*16+16).
// Streams attn_bias exactly once (non-temporal); online softmax with DPP row
// reductions; WMMA f32 for QK^T and P*V.
// ---------------------------------------------------------------------------
__global__ __launch_bounds__(256) void attn_kernel(
    const float* __restrict__ q, const float* __restrict__ kexp,
    const float* __restrict__ vexp, const float* __restrict__ bias,
    const float* __restrict__ law, const unsigned char* __restrict__ kpm,
    const unsigned char* __restrict__ em, float* __restrict__ attn)
{
  __shared__ float pbuf[8][16][18];   // per-wave probs tile, padded stride 18

  const int nTB = T_ / TBLK;          // 4
  int blk = blockIdx.x;
  int tb  = blk % nTB;
  int h   = (blk / nTB) % NH_;
  int b   = blk / (nTB * NH_);
  int wave = threadIdx.x >> 5;
  int lane = threadIdx.x & 31;
  int half = lane >> 4;               // 0 | 1
  int ln16 = lane & 15;
  int t0 = tb * TBLK + wave * 16;     // this wave's first query row

  // Q fragments in WMMA A layout: lane = row (ln16), K-pair by half.
  v2f aq[8];
  {
    const float* qrow = q + ((size_t)b * T_ + t0 + ln16) * H_ + h * D_ + 2 * half;
    #pragma unroll
    for (int kk = 0; kk < 8; ++kk) {
      v2f t = *(const v2f*)(qrow + kk * 4);
      aq[kk] = t * SCALING_;
    }
  }

  v8f o0 = {};                        // output accumulators, d-tiles 0 and 1
  v8f o1 = {};
  float rm[8], rl[8];                 // running max / sum; slot i <-> row i+8*half
  #pragma unroll
  for (int i = 0; i < 8; ++i) { rm[i] = -INFINITY; rl[i] = 0.f; }

  const float* biasBase = bias + (((size_t)b * NH_ + h) * T_ + t0) * S_;
  const float* lawBase  = law + ((size_t)b * T_ + t0) * S_;

  for (int s0 = 0; s0 < S_; s0 += 16) {
    // ---- scores tile = Q x K^T (8 WMMAs over D=32) ----
    v8f c = {};
    {
      const float* krow =
          kexp + ((size_t)b * S_ + s0 + ln16) * H_ + h * D_ + 2 * half;
      #pragma unroll
      for (int kk = 0; kk < 8; ++kk) {
        v2f bk = *(const v2f*)(krow + kk * 4);
        c = wmma_f32(aq[kk], bk, c);
      }
    }
    // key-padding / expand mask for this lane's key column
    int sc = s0 + ln16;
    bool kp = (sc < T_) ? (kpm[b * T_ + sc] != 0) : (em[b * E_ + (sc - T_)] != 0);

    // ---- bias add, masking, online softmax (per C-layout slot) ----
    #pragma unroll
    for (int i = 0; i < 8; ++i) {
      int m = i + 8 * half;
      // bias: 201 MB single-use stream -> non-temporal (keep L2 for K/V/law/W)
      float bi = __builtin_nontemporal_load(biasBase + (size_t)m * S_ + s0 + ln16);
      float lw = lawBase[(size_t)m * S_ + s0 + ln16];
      bool msk = kp || (lw <= CUTOFF_);
      float sv = msk ? -INFINITY : (c[i] + bi);
      float mx = row_max16(sv);                 // row max (16 lanes, DPP)
      float mnew = fmaxf(rm[i], mx);
      bool dead = (mnew == -INFINITY);          // whole row masked so far
      float alpha = dead ? 1.f : __expf(rm[i] - mnew);   // rm==-inf -> 0
      float p = (dead || msk) ? 0.f : __expf(sv - mnew);
      float rs = row_sum16(p);                  // softmax denom (excludes law)
      rl[i] = rl[i] * alpha + rs;
      rm[i] = mnew;
      o0[i] *= alpha;
      o1[i] *= alpha;
      pbuf[wave][m][ln16] = p * lw;             // post-softmax law reweight
    }
    __syncthreads();

    // ---- P x V (transpose probs via LDS into A layout; 8 WMMAs) ----
    #pragma unroll
    for (int kk = 0; kk < 16; kk += 4) {
      v2f pa = *(const v2f*)&pbuf[wave][ln16][kk + 2 * half];
      const float* vb =
          vexp + ((size_t)b * S_ + s0 + kk + 2 * half) * H_ + h * D_ + ln16;
      v2f b0; b0.x = vb[0];  b0.y = vb[H_];
      v2f b1; b1.x = vb[16]; b1.y = vb[H_ + 16];
      o0 = wmma_f32(pa, b0, o0);
      o1 = wmma_f32(pa, b1, o1);
    }
    __syncthreads();
  }

  // ---- normalize by softmax denominator, write attn [B,T,H] ----
  #pragma unroll
  for (int i = 0; i < 8; ++i) {
    int m = i + 8 * half;
    float inv = 1.f / rl[i];
    float* orow = attn + ((size_t)b * T_ + t0 + m) * H_ + h * D_;
    orow[ln16]      = o0[i] * inv;
    orow[16 + ln16] = o1[i] * inv;
  }
}

// ---------------------------------------------------------------------------
// Kernel 3: LayerNorm over H per (b,t) row (population variance).
// ---------------------------------------------------------------------------
__global__ __launch_bounds__(256) void ln_kernel(
    const float* __restrict__ attn, const float* __restrict__ lnw,
    const float* __restrict__ lnb, float* __restrict__ out)
{
  __shared__ float rs[8], rs2[8];
  size_t row = blockIdx.x;
  const float4* x4 = (const float4*)(attn + row * H_);
  float4 xv = x4[threadIdx.x];
  float s  = xv.x + xv.y + xv.z + xv.w;
  float s2 = xv.x * xv.x + xv.y * xv.y + xv.z * xv.z + xv.w * xv.w;
  #pragma unroll
  for (int off = 16; off >= 1; off >>= 1) {
    s  += __shfl_xor(s, off);
    s2 += __shfl_xor(s2, off);
  }
  int lane = threadIdx.x & 31, wid = threadIdx.x >> 5;
  if (lane == 0) { rs[wid] = s; rs2[wid] = s2; }
  __syncthreads();
  float tot = 0.f, tot2 = 0.f;
  #pragma unroll
  for (int i = 0; i < 8; ++i) { tot += rs[i]; tot2 += rs2[i]; }
  float mean = tot * (1.f / H_);
  float var  = tot2 * (1.f / H_) - mean * mean;
  float inv  = rsqrtf(var + EPS_);
  float4 wv = ((const float4*)lnw)[threadIdx.x];
  float4 bv = ((const float4*)lnb)[threadIdx.x];
  float4 y;
  y.x = (xv.x - mean) * inv * wv.x + bv.x;
  y.y = (xv.y - mean) * inv * wv.y + bv.y;
  y.z = (xv.z - mean) * inv * wv.z + bv.z;
  y.w = (xv.w - mean) * inv * wv.w + bv.w;
  ((float4*)(out + row * H_))[threadIdx.x] = y;
}

// ---------------------------------------------------------------------------
// Kernel 4: out = LN(2048x1024) x out_w^T via WMMA f32.
// 64x64 C-tile per WG (8 waves: mt = w&3, n-tile pair = (w>>2)*2).
// ---------------------------------------------------------------------------
__global__ __launch_bounds__(256) void proj_kernel(
    const float* __restrict__ ln, const float* __restrict__ w,
    float* __restrict__ out)
{
  __shared__ float As[64][34];
  __shared__ float Bs[64][34];
  int m0 = blockIdx.y * 64;
  int n0 = blockIdx.x * 64;
  int wave = threadIdx.x >> 5;
  int lane = threadIdx.x & 31;
  int half = lane >> 4;
  int ln16 = lane & 15;
  int mt  = wave & 3;
  int ntb = (wave >> 2) * 2;          // 0 or 2
  v8f c0 = {}, c1 = {};
  int lr = threadIdx.x >> 3;          // 0..31
  int lc = (threadIdx.x & 7) * 4;     // 0,4,...,28

  for (int kb = 0; kb < H_; kb += 32) {
    __syncthreads();
    #pragma unroll
    for (int it = 0; it < 2; ++it) {
      int r = lr + it * 32;
      *(float4*)&As[r][lc] = *(const float4*)(ln + (size_t)(m0 + r) * H_ + kb + lc);
      *(float4*)&Bs[r][lc] = *(const float4*)(w  + (size_t)(n0 + r) * H_ + kb + lc);
    }
    __syncthreads();
    #pragma unroll
    for (int kk = 0; kk < 32; kk += 4) {
      v2f a  = *(const v2f*)&As[mt * 16 + ln16][kk + 2 * half];
      v2f b0 = *(const v2f*)&Bs[ntb * 16 + ln16][kk + 2 * half];
      v2f b1 = *(const v2f*)&Bs[(ntb + 1) * 16 + ln16][kk + 2 * half];
      c0 = wmma_f32(a, b0, c0);
      c1 = wmma_f32(a, b1, c1);
    }
  }
  #pragma unroll
  for (int i = 0; i < 8; ++i) {
    int m = m0 + mt * 16 + i + 8 * half;
    int n = n0 + ntb * 16 + ln16;
    out[(size_t)m * H_ + n]      = c0[i];
    out[(size_t)m * H_ + n + 16] = c1[i];
  }
}

// ---------------------------------------------------------------------------
extern "C" void kernel_launch(void* const* d_in, const int* in_sizes, int n_in,
                              void* d_out, int out_size, void* d_ws, size_t ws_size,
                              hipStream_t stream) {
  (void)in_sizes; (void)n_in; (void)out_size; (void)ws_size;
  const float* q    = (const float*)d_in[0];
  const float* k    = (const float*)d_in[1];
  const float* v    = (const float*)d_in[2];
  const float* bias = (const float*)d_in[3];
  const float* law  = (const float*)d_in[4];
  const float* outw = (const float*)d_in[5];
  const float* lnw  = (const float*)d_in[6];
  const float* lnb  = (const float*)d_in[7];
  const unsigned char* kpm = (const unsigned char*)d_in[8];
  const int* outcell = (const int*)d_in[9];
  const unsigned char* em = (const unsigned char*)d_in[10];
  float* out = (float*)d_out;

  float* ws    = (float*)d_ws;
  float* kexp  = ws;                             // B*S*H
  float* vexp  = kexp + (size_t)B_ * S_ * H_;    // B*S*H
  float* attn  = vexp + (size_t)B_ * S_ * H_;    // B*T*H
  float* lnout = attn + (size_t)B_ * T_ * H_;    // B*T*H  (total ~42 MB)

  expand_kv_kernel<<<B_ * S_, 256, 0, stream>>>(k, v, outcell, kexp, vexp);
  attn_kernel<<<B_ * NH_ * (T_ / TBLK), 256, 0, stream>>>(
      q, kexp, vexp, bias, law, kpm, em, attn);
  ln_kernel<<<B_ * T_, 256, 0, stream>>>(attn, lnw, lnb, lnout);
  dim3 pg(H_ / 64, (B_ * T_) / 64);
  proj_kernel<<<pg, 256, 0, stream>>>(lnout, outw, out);
}